// ImageExtract_43997644980940
// MI455X (gfx1250) — compile-verified
//
#include <hip/hip_runtime.h>

// ---------------------------------------------------------------------------
// ImageExtract: out[l,b,c,ww,h] = x[b,c,2l+ww,h]
//   x: (32,32,256,32) f32, out: (112,32,32,32,32) f32
// Pure data movement (~470MB out, 32MB in). HBM-bound floor ~21us @ 23.3TB/s.
// CDNA5 Tensor Data Mover implementation: per chunk, a 2D strided tile
// (row = contiguous 4KB window, src stride 8192 elems) is DMA'd global->LDS,
// then DMA'd LDS->global (contiguous). TDM ops of a wave complete in order,
// so the store needs no explicit wait after the load.
// Store uses TH=NT (cpol=1): the 470MB write-once stream shouldn't evict the
// 32MB input working set (re-read ~16x by overlapping windows) from L2.
// ---------------------------------------------------------------------------

#define L_SEQ           112
#define ROW_ELEMS       1024u              // WIN_W * h = 32*32 contiguous f32
#define ROWS_PER_CHUNK  8u                 // bc-rows per TDM descriptor (32KB)
#define BC_ROWS         1024u              // b*c
#define CHUNKS_PER_L    (BC_ROWS / ROWS_PER_CHUNK)      // 128
#define TOTAL_CHUNKS    (L_SEQ * CHUNKS_PER_L)          // 14336
#define SRC_ROW_STRIDE  8192ull            // w*h elems between (b,c) rows
#define DST_ROW_STRIDE  1024ull            // contiguous output rows

#define CPOL_RT         0                  // default temporal
#define CPOL_NT         1                  // TH[2:0]=1 -> non-temporal

typedef unsigned int       u32;
typedef unsigned long long u64;
typedef __attribute__((ext_vector_type(4))) u32   u32x4;
typedef __attribute__((ext_vector_type(4))) int   i32x4;
typedef __attribute__((ext_vector_type(8))) int   i32x8;
typedef __attribute__((ext_vector_type(4))) float f32x4;

// D# group 0: [1:0]=count(1 valid), [63:32]=lds byte addr,
// [120:64]=global byte addr, [127:126]=type(2=image)
__device__ __forceinline__ u32x4 tdm_group0(u32 lds_off, const void* gptr) {
  u64 ga = (u64)gptr;
  u32x4 g0;
  g0.x = 1u;                                           // count=1, user mode
  g0.y = lds_off;                                      // LDS byte address
  g0.z = (u32)ga;                                      // global_addr[31:0]
  g0.w = ((u32)(ga >> 32) & 0x01FFFFFFu) | (2u << 30); // [56:32] | type=2
  return g0;
}

// D# group 1: data_size=2 (4B) at [17:16]; tensor_dim0[79:48];
// tensor_dim1[111:80]; tile_dim0[127:112]; tile_dim1[143:128];
// tensor_dim0_stride[207:160] (elems). Everything else 0.
__device__ __forceinline__ i32x8 tdm_group1(u32 tile1, u64 stride0_elems) {
  i32x8 g1;
  g1[0] = (int)(2u << 16);                                   // data_size = 4B
  g1[1] = (int)((ROW_ELEMS & 0xFFFFu) << 16);                // tdim0 lo16
  g1[2] = (int)(((ROW_ELEMS >> 16) & 0xFFFFu)
               | ((tile1 & 0xFFFFu) << 16));                 // tdim0 hi | tdim1 lo
  g1[3] = (int)(((tile1 >> 16) & 0xFFFFu)
               | (ROW_ELEMS << 16));                         // tdim1 hi | tile_dim0
  g1[4] = (int)(tile1 & 0xFFFFu);                            // tile_dim1 (tile_dim2=0)
  g1[5] = (int)(u32)(stride0_elems & 0xFFFFFFFFull);         // stride0[31:0]
  g1[6] = (int)(u32)((stride0_elems >> 32) & 0xFFFFu);       // stride0[47:32]
  g1[7] = 0;
  return g1;
}

__global__ void __launch_bounds__(32)
image_extract_tdm(const float* __restrict__ x, float* __restrict__ out) {
  __shared__ float buf[ROWS_PER_CHUNK * ROW_ELEMS];          // 32 KB staging

  const u32 chunk = blockIdx.x;                              // wave-uniform
  const u32 l     = chunk / CHUNKS_PER_L;
  const u32 r0    = (chunk % CHUNKS_PER_L) * ROWS_PER_CHUNK; // first bc-row

  // Source: rows r0..r0+7 (stride 8192 elems), each a contiguous 4KB window
  // starting at w=2l (element offset 2l*32 within the (w,h) plane).
  const float* src = x   + (u64)r0 * SRC_ROW_STRIDE + (u64)(2u * l) * 32u;
  // Destination: fully contiguous 32KB at ((l*1024 + r0) * 1024) elems.
  float*       dst = out + ((u64)l * BC_ROWS + r0) * (u64)ROW_ELEMS;

  const u32 lds_off = (u32)(u64)&buf[0];   // flat LDS addr low 32 = LDS offset

#if defined(__HIP_DEVICE_COMPILE__) && \
    __has_builtin(__builtin_amdgcn_tensor_load_to_lds) && \
    __has_builtin(__builtin_amdgcn_tensor_store_from_lds)
  u32x4 ld_g0 = tdm_group0(lds_off, src);
  i32x8 ld_g1 = tdm_group1(ROWS_PER_CHUNK, SRC_ROW_STRIDE);
  u32x4 st_g0 = tdm_group0(lds_off, dst);
  i32x8 st_g1 = tdm_group1(ROWS_PER_CHUNK, DST_ROW_STRIDE);
  i32x4 z4 = {0, 0, 0, 0};
#if __clang_major__ >= 23
  i32x8 z8 = {0, 0, 0, 0, 0, 0, 0, 0};
  __builtin_amdgcn_tensor_load_to_lds  (ld_g0, ld_g1, z4, z4, z8, CPOL_RT);
  __builtin_amdgcn_tensor_store_from_lds(st_g0, st_g1, z4, z4, z8, CPOL_NT);
#else
  __builtin_amdgcn_tensor_load_to_lds  (ld_g0, ld_g1, z4, z4, CPOL_RT);
  __builtin_amdgcn_tensor_store_from_lds(st_g0, st_g1, z4, z4, CPOL_NT);
#endif
#if __has_builtin(__builtin_amdgcn_s_wait_tensorcnt)
  __builtin_amdgcn_s_wait_tensorcnt(0);
#endif
#else
  // Fallback (no TDM builtin in this toolchain): direct vectorized copy of
  // the same 8x4KB chunk; 32 lanes, 128-bit accesses, coalesced both sides.
  (void)buf;
  const u32 lane = threadIdx.x;
  for (u32 r = 0; r < ROWS_PER_CHUNK; ++r) {
    const f32x4* s = (const f32x4*)(src + (u64)r * SRC_ROW_STRIDE);
    f32x4*       d = (f32x4*)(dst + (u64)r * DST_ROW_STRIDE);
    for (u32 j = lane; j < ROW_ELEMS / 4u; j += 32u)
      d[j] = s[j];
  }
#endif
}

extern "C" void kernel_launch(void* const* d_in, const int* in_sizes, int n_in,
                              void* d_out, int out_size, void* d_ws, size_t ws_size,
                              hipStream_t stream) {
  (void)in_sizes; (void)n_in; (void)out_size; (void)d_ws; (void)ws_size;
  const float* x = (const float*)d_in[0];
  float* out     = (float*)d_out;
  // One wave per block; each block DMAs one 32KB chunk. 14336 blocks cover
  // all (l, bc-row) chunks; ~10 blocks resident per WGP (LDS-limited) keeps
  // ~320KB of tensor DMA in flight per WGP while the VALU stays idle.
  image_extract_tdm<<<TOTAL_CHUNKS, 32, 0, stream>>>(x, out);
}